// YOLOv3Loss_9672266351325
// MI455X (gfx1250) — compile-verified
//
#include <hip/hip_runtime.h>

#define CH 85
#define CLS 80
#define RS 260   // LDS row stride (words); 260 % 64 == 4 -> bank-conflict-free GEMM loads

typedef __attribute__((ext_vector_type(2))) float v2f;
typedef __attribute__((ext_vector_type(8))) float v8f;
// 4-wide vector load with only 4-byte alignment guarantee (records are 85 floats,
// so 16B alignment is impossible; CDNA5 DWORD alignment mode handles 4B-aligned b128).
typedef float f4u __attribute__((ext_vector_type(4), aligned(4)));

__device__ __forceinline__ float bce_elem(float z, float x) {
    // max(x,0) - x*z + log1p(exp(-|x|))
    return fmaxf(x, 0.0f) - x * z + log1pf(expf(-fabsf(x)));
}

__global__ void init_ws_kernel(int* cnt) {
    if (threadIdx.x < 32) cnt[threadIdx.x] = 0;
}

// Compact valid (obj != 0) target boxes per image into a dense per-image list.
__global__ void compact_kernel(const float* __restrict__ lab, int B, int N,
                               float* __restrict__ boxes, int* __restrict__ cnt) {
    int i = blockIdx.x * blockDim.x + threadIdx.x;
    if (i >= B * N) return;
    int b = i / N;
    const float* L = lab + (size_t)i * CH;
    if (L[4] != 0.0f) {
        int slot = atomicAdd(&cnt[b], 1);
        float4* dst = (float4*)(boxes + ((size_t)b * N + slot) * 4);
        *dst = make_float4(L[0], L[1], L[2], L[3]);
    }
}

__global__ __launch_bounds__(256) void layer_kernel(
        const float* __restrict__ outp, const float* __restrict__ lab,
        int B, int g,
        float a0w, float a0h, float a1w, float a1h, float a2w, float a2h,
        const float* __restrict__ boxes, const int* __restrict__ cnt,
        float* __restrict__ blockAcc) {
    // Transposed partials: row = accumulator (16, rows 8..15 zero-padding),
    // col = thread (256). Row stride 260 words for bank-conflict-free access.
    __shared__ float partT[16 * RS];
    const int N = g * g * 3;
    const int M = B * N;
    const int i = blockIdx.x * blockDim.x + threadIdx.x;

    float p[8] = {0, 0, 0, 0, 0, 0, 0, 0};
    if (i < M) {
        int b = i / N;
        int r = i - b * N;
        int cell = r / 3;
        int a = r - cell * 3;
        int gy = cell / g;
        int gx = cell - gy * g;
        const float* F = outp + (size_t)i * CH;
        const float* L = lab + (size_t)i * CH;
        __builtin_prefetch(F, 0, 3);   // global_prefetch_b8
        __builtin_prefetch(L, 0, 3);

        // Head: [tx, ty, tw, th] as one b128 + conf as b32 (4B-aligned records).
        f4u Fh = *(const f4u*)F;
        f4u Lh = *(const f4u*)L;
        float t0 = Fh[0], t1 = Fh[1], t2 = Fh[2], t3 = Fh[3], t4 = F[4];
        float lx = Lh[0], ly = Lh[1], lw = Lh[2], lh = Lh[3], obj = L[4];
        float gf = (float)g, inp = gf * 32.0f;
        float aw = (a == 0) ? a0w : ((a == 1) ? a1w : a2w);
        float ah = (a == 0) ? a0h : ((a == 1) ? a1h : a2h);

        // xy BCE term (reference takes the mean over ALL cells)
        float zx = lx * gf - (float)gx;
        float zy = ly * gf - (float)gy;
        p[0] = bce_elem(zx, t0) + bce_elem(zy, t1);

        float scale = 2.0f - lw * lh;
        p[1] = obj * scale;

        // wh squared loss; raw_true_wh zeroed when obj == 0
        float rtw = logf(lw * inp / aw);
        float rth = logf(lh * inp / ah);
        if (obj == 0.0f) { rtw = 0.0f; rth = 0.0f; }
        float dw = rtw - t2, dh = rth - t3;
        p[2] = obj * scale * 0.5f * (dw * dw + dh * dh);

        // conf BCE term (mean over all cells)
        p[3] = bce_elem(obj, t4);

        // predicted box for the ignore mask
        float px = (1.0f / (1.0f + expf(-t0)) + (float)gx) / gf;
        float py = (1.0f / (1.0f + expf(-t1)) + (float)gy) / gf;
        float pw = expf(t2) * aw / inp;
        float ph = expf(t3) * ah / inp;
        float p_xmin = px - pw * 0.5f, p_xmax = px + pw * 0.5f;
        float p_ymin = py - ph * 0.5f, p_ymax = py + ph * 0.5f;
        float area1 = pw * ph;
        int nt = cnt[b];
        const float4* tbx = (const float4*)(boxes + (size_t)b * N * 4);
        float best = -1.0f;   // empty target set -> best = -1 (matches reference)
        for (int t = 0; t < nt; ++t) {
            float4 bx = tbx[t];
            float t_xmin = bx.x - bx.z * 0.5f, t_xmax = bx.x + bx.z * 0.5f;
            float t_ymin = bx.y - bx.w * 0.5f, t_ymax = bx.y + bx.w * 0.5f;
            // NOTE: reference uses maximum() for inter_maxes (quirk) -> replicate
            float ix0 = fmaxf(p_xmin, t_xmin);
            float iy0 = fmaxf(p_ymin, t_ymin);
            float ix1 = fmaxf(p_xmax, t_xmax);
            float iy1 = fmaxf(p_ymax, t_ymax);
            float iw = fmaxf(ix1 - ix0, 0.0f);
            float ih = fmaxf(iy1 - iy0, 0.0f);
            float inter = iw * ih;
            float area2 = bx.z * bx.w;
            float iou = inter / (area1 + area2 - inter);
            best = fmaxf(best, iou);
        }
        float ignore = (best < 0.5f) ? 1.0f : 0.0f;
        p[4] = obj + (1.0f - obj) * ignore;

        // class BCE: 80 values = 20 x b128 per record, starting at +5 (4B aligned)
        const f4u* Fc = (const f4u*)(F + 5);
        const f4u* Lc = (const f4u*)(L + 5);
        float cl = 0.0f;
#pragma unroll 4
        for (int k = 0; k < CLS / 4; ++k) {
            f4u fv = Fc[k];
            f4u lv = Lc[k];
            cl += bce_elem(lv[0], fv[0]) + bce_elem(lv[1], fv[1])
                + bce_elem(lv[2], fv[2]) + bce_elem(lv[3], fv[3]);
        }
        p[5] = cl;
        p[6] = obj;
    }

    // Cooperative transposed store (consecutive lanes -> consecutive banks),
    // rows 8..15 zeroed so the GEMM below needs no predication.
#pragma unroll
    for (int j = 0; j < 8; ++j) {
        partT[j * RS + threadIdx.x] = p[j];
        partT[(j + 8) * RS + threadIdx.x] = 0.0f;
    }
    __syncthreads();

    // Block-level reduction as a GEMM on the matrix core:
    //   D(16x16) += A(16x4) x B(4x16), A = partials (rows = accumulators,
    //   cols = 4 threads per K-step), B = ones.  After 64 K-steps,
    //   D[m][n] = sum over all 256 threads of accumulator m.
    // A-matrix VGPR layout (ISA 7.12.2): lanes 0-15 hold M=lane with K0 (v0),
    // K1 (v1); lanes 16-31 hold M=lane-16 with K2 (v0), K3 (v1) -> each lane
    // loads {part[m][t], part[m][t+1]} = one aligned ds_load_b64, unconditional.
    if (threadIdx.x < 32) {          // wave 0, EXEC all ones inside this branch
        int lane = threadIdx.x;
        int m = lane & 15;
        int koff = (lane >> 4) << 1;          // 0 for lanes 0-15, 2 for 16-31
        const float* rowp = &partT[m * RS + koff];
        v2f bones; bones[0] = 1.0f; bones[1] = 1.0f;
        v8f acc = {};
#pragma unroll 4
        for (int s = 0; s < 64; ++s) {
            v2f av = *(const v2f*)(rowp + 4 * s);   // ds_load_b64
            acc = __builtin_amdgcn_wmma_f32_16x16x4_f32(
                false, av, false, bones, (short)0, acc, false, false);
        }
        // D layout: VGPR j, lanes 0-15 -> M=j, N=lane. Lane 0 / VGPR j = total_j.
        if (lane == 0) {
            float* dst = blockAcc + (size_t)blockIdx.x * 8;
            dst[0] = acc[0]; dst[1] = acc[1]; dst[2] = acc[2]; dst[3] = acc[3];
            dst[4] = acc[4]; dst[5] = acc[5]; dst[6] = acc[6]; dst[7] = acc[7];
        }
    }
}

// Deterministic fixed-order cross-block reduction + final scalar combine.
__global__ void finalize_kernel(const float* __restrict__ blockAcc,
                                float* __restrict__ out_loss,
                                int B, int g0, int g1, int g2,
                                int nb0, int nb1, int nb2) {
    if (threadIdx.x != 0 || blockIdx.x != 0) return;
    int gs[3]  = {g0, g1, g2};
    int nbs[3] = {nb0, nb1, nb2};
    float loss = 0.0f;
    for (int l = 0; l < 3; ++l) {
        const float* ba = blockAcc + (size_t)l * 2048;
        float acc[8] = {0, 0, 0, 0, 0, 0, 0, 0};
        for (int bk = 0; bk < nbs[l]; ++bk)
            for (int j = 0; j < 7; ++j)
                acc[j] += ba[bk * 8 + j];
        float Ncells = (float)(B * gs[l] * gs[l] * 3);
        float S_xy   = acc[0] / (Ncells * 2.0f);
        float S_conf = acc[3] / Ncells;
        float S_cls  = acc[5] / (Ncells * 80.0f);
        float bf = (float)B;
        loss += S_xy * acc[1] / bf          // xy_loss
              + acc[2] / bf                 // wh_loss
              + S_conf * acc[4] / bf        // conf_loss
              + S_cls * acc[6] / bf;        // class_loss
    }
    *out_loss = loss;
}

extern "C" void kernel_launch(void* const* d_in, const int* in_sizes, int n_in,
                              void* d_out, int out_size, void* d_ws, size_t ws_size,
                              hipStream_t stream) {
    // setup_inputs() dict order: o0, l0, o1, l1, o2, l2
    const float* o[3] = {(const float*)d_in[0], (const float*)d_in[2], (const float*)d_in[4]};
    const float* l[3] = {(const float*)d_in[1], (const float*)d_in[3], (const float*)d_in[5]};
    const int B = 8;
    const int gs[3] = {13, 26, 52};
    static const float anch[3][6] = {
        {116.f, 90.f, 156.f, 198.f, 373.f, 326.f},   // mask [6,7,8]
        { 30.f, 61.f,  62.f,  45.f,  59.f, 119.f},   // mask [3,4,5]
        { 10.f, 13.f,  16.f,  30.f,  33.f,  23.f}};  // mask [0,1,2]

    int*   cnt      = (int*)d_ws;          // [3][8] per-image valid-target counters
    float* ws       = (float*)d_ws;
    float* blockAcc = ws + 32;             // 3 layers x 256 blocks x 8 floats
    size_t boxBase[3];
    size_t off = 32 + 3 * 2048;
    for (int i = 0; i < 3; ++i) {
        boxBase[i] = off;
        off += (size_t)B * gs[i] * gs[i] * 3 * 4;    // per-image box lists
    }

    init_ws_kernel<<<1, 32, 0, stream>>>(cnt);

    int nb[3];
    for (int i = 0; i < 3; ++i) {
        int N = gs[i] * gs[i] * 3;
        int M = B * N;
        nb[i] = (M + 255) / 256;
        compact_kernel<<<nb[i], 256, 0, stream>>>(l[i], B, N, ws + boxBase[i], cnt + i * 8);
    }
    for (int i = 0; i < 3; ++i) {
        layer_kernel<<<nb[i], 256, 0, stream>>>(
            o[i], l[i], B, gs[i],
            anch[i][0], anch[i][1], anch[i][2], anch[i][3], anch[i][4], anch[i][5],
            ws + boxBase[i], cnt + i * 8, blockAcc + (size_t)i * 2048);
    }
    finalize_kernel<<<1, 1, 0, stream>>>(blockAcc, (float*)d_out, B,
                                         gs[0], gs[1], gs[2], nb[0], nb[1], nb[2]);
}